// _GATModule_47012712022067
// MI455X (gfx1250) — compile-verified
//
#include <hip/hip_runtime.h>

typedef __attribute__((ext_vector_type(2))) float v2f;
typedef __attribute__((ext_vector_type(8))) float v8f;

// ---------------------------------------------------------------------------
// GEMM: Out[nrows,NCOLS] = X[nrows,128] @ W[128,NCOLS] + bias, f32 WMMA 16x16x4
// One wave per 16x16 output tile; K = 128 fixed; NCOLS compile-time so every
// global load uses an immediate offset (no 64-bit address math in the loop).
// A layout (ISA 7.12.2, 32-bit A 16x4): lanes 0-15 -> K=0(v0),1(v1); lanes 16-31 -> K=2,3
// B layout mirrored over N; C/D: VGPR r -> M=r (lanes 0-15), M=r+8 (lanes 16-31)
// ---------------------------------------------------------------------------
template <int NCOLS>
__global__ void gemm_k128_wmma(const float* __restrict__ X,
                               const float* __restrict__ W,
                               const float* __restrict__ Bias,
                               float* __restrict__ Out,
                               int nrows)
{
    const int wavesPerBlk = blockDim.x >> 5;
    constexpr int colTiles = NCOLS >> 4;
    const int totalTiles = (nrows >> 4) * colTiles;
    int t = blockIdx.x * wavesPerBlk + (threadIdx.x >> 5);
    if (t >= totalTiles) return;                  // wave-uniform: EXEC stays all-1s
    int rowTile = t / colTiles;
    int colTile = t - rowTile * colTiles;

    int lane = threadIdx.x & 31;
    int half = lane >> 4;                         // 0: K=0,1  1: K=2,3
    int l15  = lane & 15;

    // fold the lane-dependent K skew (2*half) into the bases once
    const float* xbase = X + (size_t)(rowTile * 16 + l15) * 128 + 2 * half;
    const float* wbase = W + colTile * 16 + l15 + (size_t)(2 * half) * NCOLS;

    v8f acc = {0.f, 0.f, 0.f, 0.f, 0.f, 0.f, 0.f, 0.f};
#pragma unroll
    for (int k0 = 0; k0 < 128; k0 += 4) {
        v2f a = *(const v2f*)(xbase + k0);        // A[m][k0+2h .. +1], 8B aligned
        v2f b;
        b.x = wbase[k0 * NCOLS];                  // B[k0+2h][n]
        b.y = wbase[(k0 + 1) * NCOLS];            // B[k0+2h+1][n]
        acc = __builtin_amdgcn_wmma_f32_16x16x4_f32(
            /*neg_a=*/false, a, /*neg_b=*/false, b,
            /*c_mod=*/(short)0, acc, /*reuse_a=*/false, /*reuse_b=*/false);
    }

    float bias = Bias[colTile * 16 + l15];
    float* orow = Out + (size_t)(rowTile * 16) * NCOLS + colTile * 16 + l15;
#pragma unroll
    for (int r = 0; r < 8; ++r) {
        int m = r + 8 * half;
        orow[(size_t)m * NCOLS] = acc[r] + bias;
    }
}

// ---------------------------------------------------------------------------
// Edge pass 1: e[i,h] = att_h . leaky_relu(gl[src] + gr[dst]); segment max(dst)
// One wave per edge. Monotonic-uint float max trick for atomicMax.
// ---------------------------------------------------------------------------
__device__ __forceinline__ unsigned fmap(float f) {
    unsigned u = __float_as_uint(f);
    return (u & 0x80000000u) ? ~u : (u | 0x80000000u);
}
__device__ __forceinline__ float funmap(unsigned u) {
    return (u & 0x80000000u) ? __uint_as_float(u ^ 0x80000000u)
                             : __uint_as_float(~u);
}

template <int H, int C>
__global__ void edge_scores(const int* __restrict__ ei, int E, int nE,
                            const float* __restrict__ gl,
                            const float* __restrict__ gr,
                            const float* __restrict__ att,   // [H*C]
                            float* __restrict__ e_ws,        // [nE*H]
                            unsigned* __restrict__ emax_u)   // [N*H]
{
    constexpr int CH  = H * C;
    constexpr int VPT = CH / 32;   // values per lane
    constexpr int LPH = 32 / H;    // lanes per head
    int i = blockIdx.x * (blockDim.x >> 5) + (threadIdx.x >> 5);
    if (i >= nE) return;
    int lane = threadIdx.x & 31;
    int src, dst;
    if (i < E) { src = ei[i]; dst = ei[E + i]; }
    else       { src = dst = i - E; }            // self-loop

    float partial = 0.f;
#pragma unroll
    for (int j = 0; j < VPT; ++j) {
        int ch = lane * VPT + j;
        float m = gl[(size_t)src * CH + ch] + gr[(size_t)dst * CH + ch];
        m = (m > 0.f) ? m : (0.2f * m);          // leaky_relu, slope 0.2
        partial += m * att[ch];
    }
#pragma unroll
    for (int off = 1; off < LPH; off <<= 1)
        partial += __shfl_xor(partial, off, 32);
    int head = lane / LPH;
    if ((lane & (LPH - 1)) == 0) {
        e_ws[(size_t)i * H + head] = partial;
        atomicMax(&emax_u[(size_t)dst * H + head], fmap(partial));
    }
}

// ---------------------------------------------------------------------------
// Edge pass 2: ex = exp(e - max[dst]); denom[dst] += ex (one thread per (i,h))
// ---------------------------------------------------------------------------
template <int H>
__global__ void edge_softmax_denom(const int* __restrict__ ei, int E, int nE,
                                   float* __restrict__ e_ws,
                                   const unsigned* __restrict__ emax_u,
                                   float* __restrict__ denom)
{
    int idx = blockIdx.x * blockDim.x + threadIdx.x;
    if (idx >= nE * H) return;
    int i = idx / H, h = idx - i * H;
    int dst = (i < E) ? ei[E + i] : (i - E);
    float mx = funmap(emax_u[(size_t)dst * H + h]);
    float ex = __expf(e_ws[idx] - mx);
    e_ws[idx] = ex;
    unsafeAtomicAdd(&denom[(size_t)dst * H + h], ex);
}

// ---------------------------------------------------------------------------
// Edge pass 3: acc[dst] += (ex/denom[dst]) * gl[src]   (one wave per edge)
// ---------------------------------------------------------------------------
template <int H, int C>
__global__ void edge_scatter(const int* __restrict__ ei, int E, int nE,
                             const float* __restrict__ gl,
                             const float* __restrict__ e_ws,
                             const float* __restrict__ denom,
                             float* __restrict__ acc)
{
    constexpr int CH  = H * C;
    constexpr int VPT = CH / 32;
    constexpr int LPH = 32 / H;
    int i = blockIdx.x * (blockDim.x >> 5) + (threadIdx.x >> 5);
    if (i >= nE) return;
    int lane = threadIdx.x & 31;
    int src, dst;
    if (i < E) { src = ei[i]; dst = ei[E + i]; }
    else       { src = dst = i - E; }
    int head = lane / LPH;
    float alpha = e_ws[(size_t)i * H + head] / denom[(size_t)dst * H + head];
#pragma unroll
    for (int j = 0; j < VPT; ++j) {
        int ch = lane * VPT + j;
        unsafeAtomicAdd(&acc[(size_t)dst * CH + ch],
                        alpha * gl[(size_t)src * CH + ch]);
    }
}

// ---------------------------------------------------------------------------
// Init + finalize
// ---------------------------------------------------------------------------
__global__ void init_layer(unsigned* __restrict__ emax, long long nMax,
                           float* __restrict__ denom, long long nDen,
                           float* __restrict__ acc, long long nAcc)
{
    long long stride = (long long)gridDim.x * blockDim.x;
    long long t0 = (long long)blockIdx.x * blockDim.x + threadIdx.x;
    for (long long i = t0; i < nMax; i += stride) emax[i] = 0x007FFFFFu; // mapped -inf
    for (long long i = t0; i < nDen; i += stride) denom[i] = 0.f;
    for (long long i = t0; i < nAcc; i += stride) acc[i] = 0.f;
}

__global__ void finalize_elu(float* __restrict__ h, const float* __restrict__ bias,
                             int CH, long long total)
{
    long long idx = (long long)blockIdx.x * blockDim.x + threadIdx.x;
    if (idx >= total) return;
    float v = h[idx] + bias[idx % CH];
    h[idx] = (v > 0.f) ? v : (__expf(v) - 1.f);
}

__global__ void finalize_bias(float* __restrict__ o, const float* __restrict__ bias,
                              int CH, long long total)
{
    long long idx = (long long)blockIdx.x * blockDim.x + threadIdx.x;
    if (idx >= total) return;
    o[idx] += bias[idx % CH];
}

// ---------------------------------------------------------------------------
extern "C" void kernel_launch(void* const* d_in, const int* in_sizes, int n_in,
                              void* d_out, int out_size, void* d_ws, size_t ws_size,
                              hipStream_t stream)
{
    const float* x    = (const float*)d_in[0];
    const int*   ei   = (const int*)  d_in[1];
    const float* Wl1  = (const float*)d_in[2];
    const float* bl1  = (const float*)d_in[3];
    const float* Wr1  = (const float*)d_in[4];
    const float* br1  = (const float*)d_in[5];
    const float* att1 = (const float*)d_in[6];
    const float* bias1= (const float*)d_in[7];
    const float* Wl2  = (const float*)d_in[8];
    const float* bl2  = (const float*)d_in[9];
    const float* Wr2  = (const float*)d_in[10];
    const float* br2  = (const float*)d_in[11];
    const float* att2 = (const float*)d_in[12];
    const float* bias2= (const float*)d_in[13];
    float* out = (float*)d_out;

    const int N  = in_sizes[0] / 128;  // 50000
    const int E  = in_sizes[1] / 2;    // 800000
    const int nE = E + N;              // edges + self loops

    // workspace carve-up (floats)
    float* p = (float*)d_ws;
    float* gl1  = p; p += (size_t)N * 128;
    float* gr1  = p; p += (size_t)N * 128;
    float* acc1 = p; p += (size_t)N * 128;   // becomes h after finalize_elu
    float* gl2  = p; p += (size_t)N * 32;
    float* gr2  = p; p += (size_t)N * 32;
    float* ews  = p; p += (size_t)nE * 4;    // reused (layer2 needs nE*1)
    unsigned* emax = (unsigned*)p; p += (size_t)N * 4;
    float* denom   = p;

    // ---- layer 1 ----
    {
        int tiles  = (N / 16) * (128 / 16);
        int blocks = (tiles + 7) / 8;
        gemm_k128_wmma<128><<<blocks, 256, 0, stream>>>(x, Wl1, bl1, gl1, N);
        gemm_k128_wmma<128><<<blocks, 256, 0, stream>>>(x, Wr1, br1, gr1, N);
    }
    init_layer<<<1024, 256, 0, stream>>>(emax, (long long)N * 4,
                                         denom, (long long)N * 4,
                                         acc1, (long long)N * 128);
    {
        int eblocks = (nE + 7) / 8;
        edge_scores<4, 32><<<eblocks, 256, 0, stream>>>(ei, E, nE, gl1, gr1, att1, ews, emax);
        int th = nE * 4;
        edge_softmax_denom<4><<<(th + 255) / 256, 256, 0, stream>>>(ei, E, nE, ews, emax, denom);
        edge_scatter<4, 32><<<eblocks, 256, 0, stream>>>(ei, E, nE, gl1, ews, denom, acc1);
    }
    finalize_elu<<<(int)(((long long)N * 128 + 255) / 256), 256, 0, stream>>>(
        acc1, bias1, 128, (long long)N * 128);

    // ---- layer 2 ----
    {
        int tiles  = (N / 16) * (32 / 16);
        int blocks = (tiles + 7) / 8;
        gemm_k128_wmma<32><<<blocks, 256, 0, stream>>>(acc1, Wl2, bl2, gl2, N);
        gemm_k128_wmma<32><<<blocks, 256, 0, stream>>>(acc1, Wr2, br2, gr2, N);
    }
    init_layer<<<1024, 256, 0, stream>>>(emax, (long long)N,
                                         denom, (long long)N,
                                         out, (long long)N * 32);
    {
        int eblocks = (nE + 7) / 8;
        edge_scores<1, 32><<<eblocks, 256, 0, stream>>>(ei, E, nE, gl2, gr2, att2, ews, emax);
        edge_softmax_denom<1><<<(nE + 255) / 256, 256, 0, stream>>>(ei, E, nE, ews, emax, denom);
        edge_scatter<1, 32><<<eblocks, 256, 0, stream>>>(ei, E, nE, gl2, ews, denom, out);
    }
    finalize_bias<<<(int)(((long long)N * 32 + 255) / 256), 256, 0, stream>>>(
        out, bias2, 32, (long long)N * 32);
}